// SegmentedParallelLorentzMovingWindow_22076131901908
// MI455X (gfx1250) — compile-verified
//
#include <hip/hip_runtime.h>

typedef float v2f __attribute__((ext_vector_type(2)));
typedef float v8f __attribute__((ext_vector_type(8)));

#define LOOKBACK 4096
#define SEG      32
#define DIM      256
#define NSEG     128
#define NBATCH   256
#define ROWS_TOTAL (NBATCH * NSEG)   // 32768
#define DP1      257
#define PITCH    260                 // floats per LDS row (pad: avoids phase-1 bank conflicts)
#define RPB      16                  // rows per block (one WMMA M-tile)

// Wave32 sum-reduction; result is wave-uniform, so broadcast it through
// readfirstlane to an SGPR. gfx1250 has SALU float (s_add_f32/s_fmac_f32/...),
// so the transcendental chains consuming this value scalarize and co-execute
// with VALU elementwise work.
__device__ __forceinline__ float waveSum(float v) {
    v += __shfl_xor(v, 16, 32);
    v += __shfl_xor(v,  8, 32);
    v += __shfl_xor(v,  4, 32);
    v += __shfl_xor(v,  2, 32);
    v += __shfl_xor(v,  1, 32);
    return __int_as_float(__builtin_amdgcn_readfirstlane(__float_as_int(v)));
}

extern "C" __global__ __launch_bounds__(256)
void lorentz_fused_kernel(const float* __restrict__ xt, const float* __restrict__ xc,
                          const float* __restrict__ xf, const float* __restrict__ xr,
                          const float* __restrict__ Wt, const float* __restrict__ bt,
                          const float* __restrict__ Wc, const float* __restrict__ bc,
                          const float* __restrict__ Wf, const float* __restrict__ bf,
                          const float* __restrict__ Wr, const float* __restrict__ br,
                          const float* __restrict__ es, const float* __restrict__ lw,
                          float* __restrict__ out)
{
    extern __shared__ float lds[];             // 4 * RPB * PITCH floats
    const int tid  = threadIdx.x;
    const int lane = tid & 31;
    const int wave = tid >> 5;
    const int r0   = blockIdx.x * RPB;

    // ---------------- Phase 1: z = x_seg @ W^T + b via WMMA f32 16x16x4 ----------------
    {
        const int s    = wave >> 1;                 // stream 0..3 (2 waves per stream)
        const int half = wave & 1;                  // col-tile half
        const int m    = lane & 15;                 // A row / B col within tile
        const int kb   = (lane >> 4) << 1;          // K pair base per A/B layout

        const float* X = (s == 0) ? xt : (s == 1) ? xc : (s == 2) ? xf : xr;
        const float* W = (s == 0) ? Wt : (s == 1) ? Wc : (s == 2) ? Wf : Wr;
        const float* Bv= (s == 0) ? bt : (s == 1) ? bc : (s == 2) ? bf : br;

        v2f a[8];
        #pragma unroll
        for (int st = 0; st < 8; ++st)
            a[st] = *(const v2f*)(X + (size_t)(r0 + m) * SEG + st * 4 + kb);

        #pragma unroll 1
        for (int t = 0; t < 8; ++t) {
            const int ct   = half * 8 + t;          // column tile 0..15
            const int dcol = ct * 16 + m;           // output feature index 0..255
            v8f acc = {0.f, 0.f, 0.f, 0.f, 0.f, 0.f, 0.f, 0.f};
            #pragma unroll
            for (int st = 0; st < 8; ++st) {
                v2f bvv = *(const v2f*)(W + (size_t)dcol * SEG + st * 4 + kb);
                acc = __builtin_amdgcn_wmma_f32_16x16x4_f32(
                          false, a[st], false, bvv, (short)0, acc, false, false);
            }
            const float bias  = Bv[dcol];
            const int   rbase = (lane >> 4) * 8;    // C/D layout: M = i + 8*(lane/16)
            #pragma unroll
            for (int i = 0; i < 8; ++i)
                lds[(s * RPB + rbase + i) * PITCH + 1 + dcol] = acc[i] + bias;
        }
    }
    __syncthreads();

    // ---------------- Phase 2: hyperbolic lift, h -> LDS and d_out ----------------
    const float tanhes = tanhf(es[0]);
    #pragma unroll 1
    for (int jj = 0; jj < 8; ++jj) {
        const int t   = wave + jj * 8;              // 64 tasks = 4 streams * 16 rows
        const int s   = t >> 4;
        const int row = t & 15;
        float* hrow = lds + (s * RPB + row) * PITCH;

        float v[8]; float part = 0.f;
        #pragma unroll
        for (int j = 0; j < 8; ++j) {
            float z = hrow[1 + lane + 32 * j] * tanhes;
            v[j] = z; part += z * z;
        }
        float n2   = waveSum(part);                 // uniform -> SALU chain below
        float norm = sqrtf(n2);
        float n    = fmaxf(norm, 1e-8f);
        float sc1  = fminf(n, 1.5f) / n;            // norm-clip to 1.5
        float n2b  = n2 * sc1 * sc1;
        float nb   = sqrtf(n2b);
        float ns   = fmaxf(nb, 1e-9f);
        float f    = sinhf(nb) / ns;                // expmap0 spatial factor
        float x0   = sqrtf(1.f + f * f * n2b);      // projx time component
        float fs   = f * sc1;

        float* orow = out + ((size_t)s * ROWS_TOTAL + (size_t)(r0 + row)) * DP1;
        if (lane == 0) { hrow[0] = x0; orow[0] = x0; }
        #pragma unroll
        for (int j = 0; j < 8; ++j) {
            float xrv = fs * v[j];
            hrow[1 + lane + 32 * j] = xrv;
            orow[1 + lane + 32 * j] = xrv;
        }
    }
    __syncthreads();

    // ---------------- Phase 3: Lorentz fusion (weighted Karcher mean, 5 iters) ----------------
    float l0 = lw[0], l1 = lw[1], l2 = lw[2], l3 = lw[3];
    float mx = fmaxf(fmaxf(l0, l1), fmaxf(l2, l3));
    float e0 = expf(l0 - mx), e1 = expf(l1 - mx), e2 = expf(l2 - mx), e3 = expf(l3 - mx);
    float esum = e0 + e1 + e2 + e3;
    const float wcs[4] = {e0 / esum, e1 / esum, e2 / esum, e3 / esum};

    #pragma unroll 1
    for (int rr = 0; rr < 2; ++rr) {
        const int row = wave + rr * 8;
        float p0[4], pr[4][8];
        #pragma unroll
        for (int c = 0; c < 4; ++c) {
            const float* hrow = lds + (c * RPB + row) * PITCH;
            p0[c] = hrow[0];
            #pragma unroll
            for (int j = 0; j < 8; ++j) pr[c][j] = hrow[1 + lane + 32 * j];
        }
        // p0 is wave-uniform (read by all lanes from the same LDS word)
        #pragma unroll
        for (int c = 0; c < 4; ++c)
            p0[c] = __int_as_float(__builtin_amdgcn_readfirstlane(__float_as_int(p0[c])));

        // init: mean = projx(expmap0(clip(sum_c w_c * logmap(origin, p_c), 2)[1:]))
        float wt0 = 0.f, wtr[8] = {0,0,0,0,0,0,0,0};
        #pragma unroll
        for (int c = 0; c < 4; ++c) {
            float part = 0.f;
            #pragma unroll
            for (int j = 0; j < 8; ++j) part += pr[c][j] * pr[c][j];
            float sp2   = waveSum(part);
            float alpha = fmaxf(p0[c], 1.f + 1e-7f);
            float dist  = acoshf(alpha);
            float u0    = p0[c] - alpha;
            float un    = sqrtf(fmaxf(sp2 - u0 * u0, 1e-12f));
            float k     = wcs[c] * dist / un;
            wt0 += k * u0;
            #pragma unroll
            for (int j = 0; j < 8; ++j) wtr[j] += k * pr[c][j];
        }
        float part = 0.f;
        #pragma unroll
        for (int j = 0; j < 8; ++j) part += wtr[j] * wtr[j];
        float rsp  = waveSum(part);
        float en   = sqrtf(rsp + wt0 * wt0);
        float fac  = fminf(en, 2.f) / fmaxf(en, 1e-12f);
        float sp2c = rsp * fac * fac;
        float nsp  = sqrtf(sp2c);
        float g    = sinhf(nsp) / fmaxf(nsp, 1e-9f);
        float gf   = g * fac;
        float mean_r[8];
        #pragma unroll
        for (int j = 0; j < 8; ++j) mean_r[j] = gf * wtr[j];
        float mean0 = sqrtf(1.f + g * g * sp2c);

        #pragma unroll 1
        for (int it = 0; it < 5; ++it) {
            float kc[4], ac[4], u0c[4];
            #pragma unroll
            for (int c = 0; c < 4; ++c) {
                float pd = 0.f;
                #pragma unroll
                for (int j = 0; j < 8; ++j) pd += mean_r[j] * pr[c][j];
                float dot   = waveSum(pd);
                float mink  = -mean0 * p0[c] + dot;
                float alpha = fmaxf(-mink, 1.f + 1e-7f);
                float dist  = acoshf(alpha);
                float u0    = p0[c] - alpha * mean0;
                float pu = 0.f;
                #pragma unroll
                for (int j = 0; j < 8; ++j) {
                    float u = pr[c][j] - alpha * mean_r[j];
                    pu += u * u;
                }
                float r2 = waveSum(pu);
                float un = sqrtf(fmaxf(r2 - u0 * u0, 1e-12f));
                kc[c] = wcs[c] * dist / un; ac[c] = alpha; u0c[c] = u0;
            }
            float wv0 = kc[0]*u0c[0] + kc[1]*u0c[1] + kc[2]*u0c[2] + kc[3]*u0c[3];
            float ta  = kc[0]*ac[0]  + kc[1]*ac[1]  + kc[2]*ac[2]  + kc[3]*ac[3];
            float wvr[8]; float psum = 0.f;
            #pragma unroll
            for (int j = 0; j < 8; ++j) {
                float v = kc[0]*pr[0][j] + kc[1]*pr[1][j] + kc[2]*pr[2][j] + kc[3]*pr[3][j]
                          - ta * mean_r[j];
                wvr[j] = v; psum += v * v;
            }
            float rs    = waveSum(psum);
            float en2   = sqrtf(rs + wv0 * wv0);
            float fc    = fminf(en2, 2.f) / fmaxf(en2, 1e-12f);
            float sc    = 0.1f * fc;                          // expmap(mean, 0.1 * clipped wv)
            float muu   = sc * sc * (rs - wv0 * wv0);         // mink(u,u)
            float theta = sqrtf(fmaxf(muu, 1e-12f));
            float ch    = coshf(theta);
            float shsc  = sinhf(theta) / theta * sc;          // sinh(theta)/un * 0.1*fac
            float ps2 = 0.f;
            #pragma unroll
            for (int j = 0; j < 8; ++j) {
                mean_r[j] = ch * mean_r[j] + shsc * wvr[j];
                ps2 += mean_r[j] * mean_r[j];
            }
            float r2m = waveSum(ps2);
            mean0 = sqrtf(1.f + r2m);                         // projx
        }

        float* orow = out + ((size_t)4 * ROWS_TOTAL + (size_t)(r0 + row)) * DP1;
        if (lane == 0) orow[0] = mean0;
        #pragma unroll
        for (int j = 0; j < 8; ++j) orow[1 + lane + 32 * j] = mean_r[j];
    }
}

extern "C" void kernel_launch(void* const* d_in, const int* in_sizes, int n_in,
                              void* d_out, int out_size, void* d_ws, size_t ws_size,
                              hipStream_t stream) {
    (void)in_sizes; (void)n_in; (void)out_size; (void)d_ws; (void)ws_size;
    const size_t smem = (size_t)4 * RPB * PITCH * sizeof(float);   // 66,560 B
    dim3 grid(ROWS_TOTAL / RPB);                                    // 2048 blocks
    lorentz_fused_kernel<<<grid, 256, smem, stream>>>(
        (const float*)d_in[0],  (const float*)d_in[1],
        (const float*)d_in[2],  (const float*)d_in[3],
        (const float*)d_in[4],  (const float*)d_in[5],
        (const float*)d_in[6],  (const float*)d_in[7],
        (const float*)d_in[8],  (const float*)d_in[9],
        (const float*)d_in[10], (const float*)d_in[11],
        (const float*)d_in[12], (const float*)d_in[13],
        (float*)d_out);
}